// FFBGLocal_18021682774971
// MI455X (gfx1250) — compile-verified
//
#include <hip/hip_runtime.h>
#include <hip/hip_bf16.h>
#include <math.h>

// ---------------------------------------------------------------------------
// Problem constants (from reference setup_inputs)
// ---------------------------------------------------------------------------
#define BATCH   8
#define CI      256
#define HWSZ    1600          // 40*40
#define DDIM    128
#define NCLOUD  32768
#define NHEAD   8
#define DH      16            // DDIM / NHEAD
#define GEM_EPS 1e-6f
#define L2_EPS  1e-12f

typedef float v2f __attribute__((ext_vector_type(2)));
typedef float v8f __attribute__((ext_vector_type(8)));
typedef int   v4i __attribute__((vector_size(16)));   // builtin's pointee type

#define GLB_AS __attribute__((address_space(1)))
#define LDS_AS __attribute__((address_space(3)))

// CDNA5 async global->LDS copy path (ASYNCcnt-tracked). Guarded so the file
// still compiles (host pass + older toolchains) with a plain-load fallback.
#if defined(__AMDGCN__) && __has_builtin(__builtin_amdgcn_global_load_async_to_lds_b128)
#define USE_ASYNC_LDS 1
#else
#define USE_ASYNC_LDS 0
#endif

#if USE_ASYNC_LDS
template <int N>
__device__ __forceinline__ void async_wait() {
#if __has_builtin(__builtin_amdgcn_s_wait_asynccnt)
  __builtin_amdgcn_s_wait_asynccnt(N);
#else
  asm volatile("s_wait_asynccnt %0" ::"i"(N) : "memory");
#endif
  asm volatile("" ::: "memory");   // keep LDS reads below the wait
}
#endif

// fp32 WMMA: D(16x16,f32) = A(16x4,f32) * B(4x16,f32) + C
__device__ __forceinline__ v8f wmma4(v2f a, v2f b, v8f c) {
  return __builtin_amdgcn_wmma_f32_16x16x4_f32(false, a, false, b, (short)0, c,
                                               false, false);
}

// Row reductions inside a 16-lane half (C/D layout: a row's 16 cols live in
// one 16-lane half of a wave32).
__device__ __forceinline__ float red16_max(float v) {
  v = fmaxf(v, __shfl_xor(v, 1, 32));
  v = fmaxf(v, __shfl_xor(v, 2, 32));
  v = fmaxf(v, __shfl_xor(v, 4, 32));
  v = fmaxf(v, __shfl_xor(v, 8, 32));
  return v;
}
__device__ __forceinline__ float red16_sum(float v) {
  v += __shfl_xor(v, 1, 32);
  v += __shfl_xor(v, 2, 32);
  v += __shfl_xor(v, 4, 32);
  v += __shfl_xor(v, 8, 32);
  return v;
}
__device__ __forceinline__ float red32_sum(float v) {
  v += __shfl_xor(v, 1, 32);
  v += __shfl_xor(v, 2, 32);
  v += __shfl_xor(v, 4, 32);
  v += __shfl_xor(v, 8, 32);
  v += __shfl_xor(v, 16, 32);
  return v;
}

// ---------------------------------------------------------------------------
// Kernel 0: zero the accumulator region of the workspace
// ---------------------------------------------------------------------------
__global__ void k_zero(float* p, int n) {
  int i = blockIdx.x * blockDim.x + threadIdx.x;
  if (i < n) p[i] = 0.0f;
}

// ---------------------------------------------------------------------------
// Kernel 1: tok = l2norm(image_feat^T @ W1 + b1)
// One wave per 16-row M-tile; computes the full 16x128 row block.
// ---------------------------------------------------------------------------
__global__ void __launch_bounds__(32)
k_gemm1_l2norm(const float* __restrict__ img, const float* __restrict__ W1,
               const float* __restrict__ b1, float* __restrict__ tok) {
  const int lane = threadIdx.x;
  const int lo = lane & 15, hi = lane >> 4;
  const int mBase = blockIdx.x * 16;          // 800 tiles, never crosses batch
  const int b  = mBase / HWSZ;
  const int hw = (mBase % HWSZ) + lo;
  const float* aBase = img + (size_t)b * CI * HWSZ + hw;

  v8f acc[8];
#pragma unroll
  for (int nt = 0; nt < 8; ++nt) acc[nt] = (v8f){0, 0, 0, 0, 0, 0, 0, 0};

  for (int k0 = 0; k0 < CI; k0 += 4) {
    const int ka = k0 + hi * 2;               // K per A/B fragment lane-half
    v2f a;
    a.x = aBase[(size_t)(ka + 0) * HWSZ];
    a.y = aBase[(size_t)(ka + 1) * HWSZ];
#pragma unroll
    for (int nt = 0; nt < 8; ++nt) {
      const int d = nt * 16 + lo;
      v2f bb;
      bb.x = W1[(ka + 0) * DDIM + d];
      bb.y = W1[(ka + 1) * DDIM + d];
      acc[nt] = wmma4(a, bb, acc[nt]);
    }
  }

  float bias[8];
#pragma unroll
  for (int nt = 0; nt < 8; ++nt) bias[nt] = b1[nt * 16 + lo];

#pragma unroll
  for (int r = 0; r < 8; ++r) {
    float ss = 0.0f;
#pragma unroll
    for (int nt = 0; nt < 8; ++nt) {
      float v = acc[nt][r] + bias[nt];
      acc[nt][r] = v;
      ss += v * v;
    }
    ss = red16_sum(ss);                       // per-row sum of squares
    const float inv = 1.0f / fmaxf(sqrtf(ss), L2_EPS);
    const int row = mBase + r + hi * 8;
#pragma unroll
    for (int nt = 0; nt < 8; ++nt)
      tok[(size_t)row * DDIM + nt * 16 + lo] = acc[nt][r] * inv;
  }
}

// ---------------------------------------------------------------------------
// Kernel 2: flash self-attention per (batch, head, 16-query tile).
// q = k = v = tok[:, head*16 : head*16+16].  Online softmax; P transposed
// C-layout -> A-layout through a 16x17 LDS tile.  Key/value tiles are staged
// to LDS with double-buffered GLOBAL_LOAD_ASYNC_TO_LDS_B128 (ASYNCcnt),
// overlapping the copy of tile kt+1 with the WMMA/softmax work on tile kt.
// ---------------------------------------------------------------------------
#define KVSTRIDE 20   // floats; 80B rows: 16B-aligned B128 chunks, no bank
                      // conflicts on column reads (5*dr % 16 != 0, dr=1..15)

#if USE_ASYNC_LDS
__device__ __forceinline__ void issue_kv(const float* __restrict__ tok,
                                         float* kvbuf, int kRow0, int dcol,
                                         int lane) {
  const int row = lane >> 2;                  // 0..7
  const int sub = (lane & 3) * 4;             // float offset in row
  const float* g0 = tok + (size_t)(kRow0 + row) * DDIM + dcol + sub;
  const float* g1 = g0 + (size_t)8 * DDIM;
  __builtin_amdgcn_global_load_async_to_lds_b128(
      (v4i*)g0, (LDS_AS v4i*)&kvbuf[row * KVSTRIDE + sub], 0, 0);
  __builtin_amdgcn_global_load_async_to_lds_b128(
      (v4i*)g1, (LDS_AS v4i*)&kvbuf[(row + 8) * KVSTRIDE + sub], 0, 0);
}
#endif

__global__ void __launch_bounds__(32)
k_attn(const float* __restrict__ tok, float* __restrict__ out) {
  __shared__ float pt[16 * 17];               // P transpose tile
#if USE_ASYNC_LDS
  __shared__ float kv[2][16 * KVSTRIDE];      // double-buffered K/V tile
#endif
  const int lane = threadIdx.x;
  const int lo = lane & 15, hi = lane >> 4;

  int idx = blockIdx.x;                       // 8*8*100 blocks
  const int qt = idx % 100; idx /= 100;
  const int h  = idx % NHEAD;
  const int b  = idx / NHEAD;
  const int dcol  = h * DH;
  const int qRow0 = b * HWSZ + qt * 16;
  const int bRow0 = b * HWSZ;

  // Q in A-layout: 4 chunks of 16x4
  v2f qa[4];
#pragma unroll
  for (int kk = 0; kk < 4; ++kk) {
    const float* p = tok + (size_t)(qRow0 + lo) * DDIM + dcol + kk * 4 + hi * 2;
    qa[kk].x = p[0];
    qa[kk].y = p[1];
  }

  v8f o = (v8f){0, 0, 0, 0, 0, 0, 0, 0};
  float mrun[8], lrun[8];
#pragma unroll
  for (int r = 0; r < 8; ++r) { mrun[r] = -INFINITY; lrun[r] = 0.0f; }

#if USE_ASYNC_LDS
  issue_kv(tok, kv[0], bRow0, dcol, lane);    // prime the pipeline
#endif

  for (int kt = 0; kt < 100; ++kt) {
    const int kRow0 = bRow0 + kt * 16;
#if USE_ASYNC_LDS
    float* kvc = kv[kt & 1];
    if (kt + 1 < 100) {                       // overlap next copy with compute
      issue_kv(tok, kv[(kt + 1) & 1], kRow0 + 16, dcol, lane);
      async_wait<2>();                        // in-order: older tile complete
    } else {
      async_wait<0>();
    }
#else
    if (kt + 1 < 100)                         // CDNA5 global_prefetch_b8
      __builtin_prefetch(tok + (size_t)(kRow0 + 16 + lo) * DDIM + dcol, 0, 0);
#endif

    // S = Q * K^T   (contraction over dh=16 -> 4 fp32 WMMAs)
    v8f S = (v8f){0, 0, 0, 0, 0, 0, 0, 0};
#pragma unroll
    for (int kk = 0; kk < 4; ++kk) {
      const int kidx = kk * 4 + hi * 2;
      v2f bs;                                  // B[K=d, N=key] = tok[key, d]
#if USE_ASYNC_LDS
      bs.x = kvc[lo * KVSTRIDE + kidx + 0];
      bs.y = kvc[lo * KVSTRIDE + kidx + 1];
#else
      const float* p = tok + (size_t)(kRow0 + lo) * DDIM + dcol + kidx;
      bs.x = p[0];
      bs.y = p[1];
#endif
      S = wmma4(qa[kk], bs, S);
    }

    // online softmax (row = one 16-lane half of each C VGPR)
#pragma unroll
    for (int r = 0; r < 8; ++r) {
      const float s    = S[r] * 0.25f;         // dh^-0.5
      const float tmax = red16_max(s);
      const float mnew = fmaxf(mrun[r], tmax);
      const float corr = __expf(mrun[r] - mnew);
      const float prob = __expf(s - mnew);
      lrun[r] = lrun[r] * corr + red16_sum(prob);
      mrun[r] = mnew;
      o[r] *= corr;
      pt[(r + hi * 8) * 17 + lo] = prob;       // P[row, key] for transpose
    }
    __syncthreads();

    // O += P * V   (contraction over 16 keys -> 4 fp32 WMMAs)
#pragma unroll
    for (int kk = 0; kk < 4; ++kk) {
      const int kidx = kk * 4 + hi * 2;
      v2f pa;                                  // A[M=row, K=key] from LDS
      pa.x = pt[lo * 17 + kidx + 0];
      pa.y = pt[lo * 17 + kidx + 1];
      v2f bo;                                  // B[K=key, N=d] = tok[key, d]
#if USE_ASYNC_LDS
      bo.x = kvc[(kidx + 0) * KVSTRIDE + lo];
      bo.y = kvc[(kidx + 1) * KVSTRIDE + lo];
#else
      const float* p = tok + (size_t)(kRow0 + kidx) * DDIM + dcol + lo;
      bo.x = p[0];
      bo.y = p[DDIM];
#endif
      o = wmma4(pa, bo, o);
    }
    __syncthreads();                           // protect pt for next tile
  }

#pragma unroll
  for (int r = 0; r < 8; ++r) {
    const int row = qRow0 + r + hi * 8;
    out[(size_t)row * DDIM + dcol + lo] = o[r] / lrun[r];
  }
}

// ---------------------------------------------------------------------------
// Kernel 3: img_out = (attn_out @ W2 + b2), stored as [B,Ci,H,W], fused with
// GeM clip^3 partial accumulation.  One wave per 16x16 output tile.
// ---------------------------------------------------------------------------
__global__ void __launch_bounds__(32)
k_gemm2(const float* __restrict__ x, const float* __restrict__ W2,
        const float* __restrict__ b2, float* __restrict__ img_out,
        float* __restrict__ gem_acc) {
  __shared__ float lds[16 * 17];
  const int lane = threadIdx.x;
  const int lo = lane & 15, hi = lane >> 4;
  const int mBase = blockIdx.x * 16;           // 800
  const int nBase = blockIdx.y * 16;           // 16
  const int b = mBase / HWSZ, hw0 = mBase % HWSZ;

  v8f acc = (v8f){0, 0, 0, 0, 0, 0, 0, 0};
  const float* aBase = x + (size_t)(mBase + lo) * DDIM;
  for (int k0 = 0; k0 < DDIM; k0 += 4) {
    const int ka = k0 + hi * 2;
    v2f a;
    a.x = aBase[ka + 0];
    a.y = aBase[ka + 1];
    v2f bb;
    bb.x = W2[(ka + 0) * CI + nBase + lo];
    bb.y = W2[(ka + 1) * CI + nBase + lo];
    acc = wmma4(a, bb, acc);
  }

  const float bias = b2[nBase + lo];
  float gsum = 0.0f;
#pragma unroll
  for (int r = 0; r < 8; ++r) {
    const float v = acc[r] + bias;
    lds[(r + hi * 8) * 17 + lo] = v;
    const float c = fmaxf(v, GEM_EPS);
    gsum += c * c * c;                         // clip^p, p=3
  }
  gsum += __shfl_xor(gsum, 16, 32);            // combine both 8-row halves
  __syncthreads();

  // transposed store: lane -> (channel = lane>>1, 8 consecutive hw)
  {
    const int c_local = lane >> 1;
    const int rhalf = (lane & 1) * 8;
    float* dst = img_out + ((size_t)(b * CI + nBase + c_local) * HWSZ) + hw0 + rhalf;
#pragma unroll
    for (int i = 0; i < 8; ++i) dst[i] = lds[(rhalf + i) * 17 + c_local];
  }
  if (lane < 16) atomicAdd(&gem_acc[b * CI + nBase + lane], gsum);
}

// ---------------------------------------------------------------------------
// Kernel 4: cloud branch: l2norm rows, clip^3, LDS-staged segment sums.
// One wave per row (128 elems = float4 per lane).
// ---------------------------------------------------------------------------
__global__ void __launch_bounds__(256)
k_cloud(const float* __restrict__ cf, const int* __restrict__ bids,
        float* __restrict__ cout, float* __restrict__ seg_acc,
        float* __restrict__ cnt_acc) {
  __shared__ float seg[BATCH * DDIM];
  __shared__ float cnt[BATCH];
  const int t = threadIdx.x;
  const int lane = t & 31;
  const int wave = t >> 5;
  for (int i = t; i < BATCH * DDIM; i += 256) seg[i] = 0.0f;
  if (t < BATCH) cnt[t] = 0.0f;
  __syncthreads();

  const int wgid = blockIdx.x * 8 + wave;
  const int nw = gridDim.x * 8;
  for (int row = wgid; row < NCLOUD; row += nw) {
    const float4 v = ((const float4*)(cf + (size_t)row * DDIM))[lane];
    float ss = v.x * v.x + v.y * v.y + v.z * v.z + v.w * v.w;
    ss = red32_sum(ss);
    const float inv = 1.0f / fmaxf(sqrtf(ss), L2_EPS);
    float4 o;
    o.x = v.x * inv; o.y = v.y * inv; o.z = v.z * inv; o.w = v.w * inv;
    ((float4*)(cout + (size_t)row * DDIM))[lane] = o;

    const int bid = bids[row];
    float cx = fmaxf(o.x, GEM_EPS), cy = fmaxf(o.y, GEM_EPS);
    float cz = fmaxf(o.z, GEM_EPS), cw = fmaxf(o.w, GEM_EPS);
    float* sp = seg + bid * DDIM + lane * 4;
    atomicAdd(sp + 0, cx * cx * cx);
    atomicAdd(sp + 1, cy * cy * cy);
    atomicAdd(sp + 2, cz * cz * cz);
    atomicAdd(sp + 3, cw * cw * cw);
    if (lane == 0) atomicAdd(&cnt[bid], 1.0f);
  }
  __syncthreads();
  for (int i = t; i < BATCH * DDIM; i += 256) atomicAdd(&seg_acc[i], seg[i]);
  if (t < BATCH) atomicAdd(&cnt_acc[t], cnt[t]);
}

// ---------------------------------------------------------------------------
// Kernel 5: finalize GeM pools
// ---------------------------------------------------------------------------
__global__ void k_finalize(const float* __restrict__ img_acc,
                           const float* __restrict__ seg_acc,
                           const float* __restrict__ cnt_acc,
                           float* __restrict__ image_gem,
                           float* __restrict__ cloud_gem) {
  const int i = blockIdx.x * blockDim.x + threadIdx.x;
  if (i < BATCH * CI) image_gem[i] = cbrtf(img_acc[i] * (1.0f / (float)HWSZ));
  if (i < BATCH * DDIM) {
    const int b = i / DDIM;
    cloud_gem[i] = cbrtf(seg_acc[i] / fmaxf(cnt_acc[b], 1.0f));
  }
}

// ---------------------------------------------------------------------------
// Host launcher
// ---------------------------------------------------------------------------
extern "C" void kernel_launch(void* const* d_in, const int* in_sizes, int n_in,
                              void* d_out, int out_size, void* d_ws, size_t ws_size,
                              hipStream_t stream) {
  const float* image_feat = (const float*)d_in[0];   // [8,256,40,40]
  const float* cloud_feat = (const float*)d_in[1];   // [32768,128]
  const int*   cloud_bids = (const int*)d_in[2];     // [32768]
  const float* W1 = (const float*)d_in[3];           // [256,128]
  const float* b1 = (const float*)d_in[4];           // [128]
  const float* W2 = (const float*)d_in[5];           // [128,256]
  const float* b2 = (const float*)d_in[6];           // [256]

  // Output layout (floats, concatenated in reference return order)
  float* out_img   = (float*)d_out;                         // 8*256*1600
  float* out_cloud = out_img + (size_t)BATCH * CI * HWSZ;   // 32768*128
  float* out_igem  = out_cloud + (size_t)NCLOUD * DDIM;     // 8*256
  float* out_cgem  = out_igem + BATCH * CI;                 // 8*128

  // Workspace layout (~13.1 MB of floats)
  float* ws      = (float*)d_ws;
  float* tok     = ws;                                      // 12800*128
  float* attn    = tok + (size_t)12800 * DDIM;              // 12800*128
  float* img_acc = attn + (size_t)12800 * DDIM;             // 2048
  float* seg_acc = img_acc + BATCH * CI;                    // 1024
  float* cnt_acc = seg_acc + BATCH * DDIM;                  // 8

  // 0) zero accumulators (ws is poisoned; must be deterministic every call)
  const int nzero = BATCH * CI + BATCH * DDIM + BATCH;      // 3080
  k_zero<<<(nzero + 255) / 256, 256, 0, stream>>>(img_acc, nzero);

  // 1) tok = l2norm(image^T @ W1 + b1)
  k_gemm1_l2norm<<<12800 / 16, 32, 0, stream>>>(image_feat, W1, b1, tok);

  // 2) flash self-attention per (b, head, q-tile)
  k_attn<<<BATCH * NHEAD * (HWSZ / 16), 32, 0, stream>>>(tok, attn);

  // 3) img_out = attn @ W2 + b2 (NCHW) + GeM accumulation
  k_gemm2<<<dim3(12800 / 16, CI / 16), 32, 0, stream>>>(attn, W2, b2,
                                                        out_img, img_acc);

  // 4) cloud branch
  k_cloud<<<256, 256, 0, stream>>>(cloud_feat, cloud_bids, out_cloud,
                                   seg_acc, cnt_acc);

  // 5) finalize GeM pools
  k_finalize<<<(BATCH * CI + 255) / 256, 256, 0, stream>>>(
      img_acc, seg_acc, cnt_acc, out_igem, out_cgem);
}